// EnhancedGNN_48095043781198
// MI455X (gfx1250) — compile-verified
//
#include <hip/hip_runtime.h>
#include <cmath>

// ---------------------------------------------------------------------------
// EnhancedGNN (2-layer GCN, 128 -> 64 -> 64 -> 1) for gfx1250 / MI455X.
//
// Phases:
//  1) deg[i]   = sum over edges of (dst==i), via global_atomic_add_f32
//  2) dis[i]   = rsqrt(deg[i] + 1)                       (self-loop)
//  3) P = X @ W   -- fp32 WMMA (v_wmma_f32_16x16x4_f32); W staged in LDS in a
//     pair-interleaved layout so each B fragment is ONE ds_load_b64 (no packing
//     moves between LDS and the WMMAs).
//  4) AGG[dst] += dis[src]*dis[dst] * P[src]             -- wave per edge,
//     2 features/lane, f32 atomics (L2-resident: P/AGG are 25.6MB << 192MB L2)
//  5) AGG = act(AGG + dis^2 * P + b)                      (in place)
//  6) out[i] = sigmoid(dot(H2[i,:], Wl)) (bl unused)             -- wave per node
// ---------------------------------------------------------------------------

typedef __attribute__((ext_vector_type(2))) float v2f;
typedef __attribute__((ext_vector_type(8))) float v8f;

#define TB 256
#define WAVES_PER_BLOCK 8   // 256 threads / wave32

// D = A(16x4,f32) * B(4x16,f32) + C(16x16,f32)
__device__ __forceinline__ v8f wmma_f32(v2f a, v2f b, v8f c) {
  return __builtin_amdgcn_wmma_f32_16x16x4_f32(
      /*neg_a=*/false, a, /*neg_b=*/false, b,
      /*c_mod=*/(short)0, c, /*reuse_a=*/false, /*reuse_b=*/false);
}

// ---- phase 1: degree via atomics (deg buffer pre-zeroed with memset) -------
__global__ void deg_kernel(const long long* __restrict__ ei, long long E,
                           float* __restrict__ deg) {
  long long e = (long long)blockIdx.x * blockDim.x + threadIdx.x;
  if (e < E) atomicAdd(&deg[ei[E + e]], 1.0f);
}

// ---- phase 2: dis = rsqrt(deg + 1), in place -------------------------------
__global__ void rsqrt_kernel(float* __restrict__ d, int n) {
  int i = blockIdx.x * blockDim.x + threadIdx.x;
  if (i < n) d[i] = rsqrtf(d[i] + 1.0f);
}

// ---- phase 3: P[N,64] = X[N,K] @ W[K,64] via fp32 WMMA ---------------------
// One wave computes a 16x64 stripe (4 N-tiles reuse one A fragment).
// A layout (32-bit A 16x4): lane L -> row M = L%16; VGPR0/1 hold K = kOff,kOff+1
//   with kOff = (L>=16) ? 2 : 0.
// B layout (4x16): VGPR0 lane L -> B[kOff][L%16], VGPR1 -> B[kOff+1][L%16].
// C/D layout: VGPR r, lane L -> (M = r + 8*(L/16), N = L%16).
//
// LDS holds W pair-interleaved: sW[(k>>1)*128 + 2*c + (k&1)] = W[k][c], so a
// B fragment {W[k][c], W[k+1][c]} is one aligned 8-byte LDS load.
template <int K>
__global__ void gemm_wmma_kernel(const float* __restrict__ X,
                                 const float* __restrict__ W,
                                 float* __restrict__ P, int N, int nTiles) {
  __shared__ float sW[K * 64];
  for (int i = threadIdx.x; i < K * 64; i += TB) {
    int k = i >> 6, c = i & 63;
    sW[(k >> 1) * 128 + (c << 1) + (k & 1)] = W[i];
  }
  __syncthreads();

  const int wave = threadIdx.x >> 5;
  const int lane = threadIdx.x & 31;
  const int tile = blockIdx.x * WAVES_PER_BLOCK + wave;
  if (tile >= nTiles) return;   // wave-uniform exit: EXEC all-1s around WMMA

  const int col  = lane & 15;            // M (for A) and N (for B/C)
  const int kOff = (lane >> 4) << 1;     // 0 or 2
  int rowM = (tile << 4) + col;
  if (rowM >= N) rowM = N - 1;           // clamp (N is a multiple of 16 here)
  const float* xrow = X + (size_t)rowM * K;

  v8f c0 = {}, c1 = {}, c2 = {}, c3 = {};
  for (int k0 = 0; k0 < K; k0 += 4) {
    v2f a = *(const v2f*)(xrow + k0 + kOff);                 // global_load_b64
    const v2f* wb = (const v2f*)&sW[((k0 + kOff) >> 1) * 128 + (col << 1)];
    v2f b0 = wb[0];                                          // ds_load_b64 each
    v2f b1 = wb[16];
    v2f b2 = wb[32];
    v2f b3 = wb[48];
    c0 = wmma_f32(a, b0, c0);
    c1 = wmma_f32(a, b1, c1);
    c2 = wmma_f32(a, b2, c2);
    c3 = wmma_f32(a, b3, c3);
  }

  const int mBase = (tile << 4) + ((lane >> 4) << 3);  // +0 or +8
  for (int r = 0; r < 8; ++r) {
    int m = mBase + r;
    if (m < N) {
      float* prow = P + (size_t)m * 64 + col;
      prow[0]  = c0[r];
      prow[16] = c1[r];
      prow[32] = c2[r];
      prow[48] = c3[r];
    }
  }
}

// ---- phase 4: edge aggregation, one wave per edge --------------------------
__global__ void agg_kernel(const float* __restrict__ P,
                           const float* __restrict__ dis,
                           const long long* __restrict__ ei, long long E,
                           float* __restrict__ AGG) {
  long long w = (long long)blockIdx.x * WAVES_PER_BLOCK + (threadIdx.x >> 5);
  const int lane = threadIdx.x & 31;
  if (w >= E) return;
  const long long s = ei[w];
  const long long d = ei[E + w];
  const float nrm = dis[s] * dis[d];
  v2f v = *(const v2f*)&P[(size_t)s * 64 + lane * 2];   // coalesced row gather
  float* ap = &AGG[(size_t)d * 64 + lane * 2];
  atomicAdd(ap,     nrm * v.x);                          // global_atomic_add_f32
  atomicAdd(ap + 1, nrm * v.y);
}

// ---- phase 5: self-loop + bias (+ relu), in place in AGG -------------------
__global__ void selfloop_kernel(float* __restrict__ AGG,
                                const float* __restrict__ P,
                                const float* __restrict__ dis,
                                const float* __restrict__ b,
                                int total, int doRelu) {
  int idx = blockIdx.x * blockDim.x + threadIdx.x;
  if (idx >= total) return;
  int i = idx >> 6, f = idx & 63;
  float di = dis[i];
  float v = AGG[idx] + di * di * P[idx] + b[f];
  AGG[idx] = doRelu ? fmaxf(v, 0.0f) : v;
}

// ---- phase 6: out = sigmoid(H @ Wl), bl unused, one wave per node ----------------
__global__ void final_kernel(const float* __restrict__ H,
                             const float* __restrict__ Wl,
                             const float* __restrict__ bl,
                             float* __restrict__ out, int n) {
  int node = blockIdx.x * WAVES_PER_BLOCK + (threadIdx.x >> 5);
  int lane = threadIdx.x & 31;
  if (node >= n) return;
  v2f h = *(const v2f*)&H[(size_t)node * 64 + lane * 2];
  v2f w = *(const v2f*)&Wl[lane * 2];
  float p = h.x * w.x + h.y * w.y;
  for (int off = 16; off > 0; off >>= 1) p += __shfl_down(p, off);
  if (lane == 0) out[node] = 1.0f / (1.0f + expf(-p));
}

extern "C" void kernel_launch(void* const* d_in, const int* in_sizes, int n_in,
                              void* d_out, int out_size, void* d_ws, size_t ws_size,
                              hipStream_t stream) {
  const float*     x  = (const float*)d_in[0];      // [N,128]
  const long long* ei = (const long long*)d_in[1];  // [2,E] int64
  const float*     W1 = (const float*)d_in[2];      // [128,64]
  const float*     b1 = (const float*)d_in[3];      // [64]
  const float*     W2 = (const float*)d_in[4];      // [64,64]
  const float*     b2 = (const float*)d_in[5];      // [64]
  const float*     Wl = (const float*)d_in[6];      // [64,1]
  const float*     bl = (const float*)d_in[7];      // [1]
  float* out = (float*)d_out;

  const int       N = in_sizes[0] / 128;
  const long long E = in_sizes[1] / 2;

  // workspace: dis[N] | P[N*64] | AGG[N*64]  -> N*129*4 bytes (~51.6 MB)
  float* dis = (float*)d_ws;
  float* P   = dis + N;
  float* AGG = P + (size_t)N * 64;

  const int nTiles     = (N + 15) / 16;
  const int gemmBlocks = (nTiles + WAVES_PER_BLOCK - 1) / WAVES_PER_BLOCK;
  const int total      = N * 64;
  const int edgeBlocks = (int)((E + TB - 1) / TB);
  const int aggBlocks  = (int)((E + WAVES_PER_BLOCK - 1) / WAVES_PER_BLOCK);

  // degrees + normalization (shared across both layers)
  hipMemsetAsync(dis, 0, sizeof(float) * (size_t)N, stream);
  deg_kernel<<<edgeBlocks, TB, 0, stream>>>(ei, E, dis);
  rsqrt_kernel<<<(N + TB - 1) / TB, TB, 0, stream>>>(dis, N);

  // ---- layer 1: h1 = relu(norm_agg(X@W1) + b1) -> AGG
  gemm_wmma_kernel<128><<<gemmBlocks, TB, 0, stream>>>(x, W1, P, N, nTiles);
  hipMemsetAsync(AGG, 0, sizeof(float) * (size_t)N * 64, stream);
  agg_kernel<<<aggBlocks, TB, 0, stream>>>(P, dis, ei, E, AGG);
  selfloop_kernel<<<(total + TB - 1) / TB, TB, 0, stream>>>(AGG, P, dis, b1, total, 1);

  // ---- layer 2: h2 = norm_agg(h1@W2) + b2 -> AGG (gemm consumes AGG first)
  gemm_wmma_kernel<64><<<gemmBlocks, TB, 0, stream>>>(AGG, W2, P, N, nTiles);
  hipMemsetAsync(AGG, 0, sizeof(float) * (size_t)N * 64, stream);
  agg_kernel<<<aggBlocks, TB, 0, stream>>>(P, dis, ei, E, AGG);
  selfloop_kernel<<<(total + TB - 1) / TB, TB, 0, stream>>>(AGG, P, dis, b2, total, 0);

  // ---- head: sigmoid(h2 @ Wl), bl not added
  final_kernel<<<(N + WAVES_PER_BLOCK - 1) / WAVES_PER_BLOCK, TB, 0, stream>>>(
      AGG, Wl, bl, out, N);
}